// VQVAE_39908836114666
// MI455X (gfx1250) — compile-verified
//
#include <hip/hip_runtime.h>

// VQ-VAE codebook quantization for MI455X (gfx1250, wave32).
//
// Shapes (hardcoded to the reference's setup_inputs):
//   x:        (B=4096, 1024) f32, viewed as (B, C=128, D=8)
//   codebook: (C=128, K=256, D=8) f32
// Outputs (concatenated in d_out, f32):
//   cw_embed: (4096, 1024)           = 16 MB
//   one_hot:  (4096, 128, 256)       = 512 MB   <-- dominates: store-bound kernel
//
// One wave32 handles a (16 batch rows x 1 code group) tile.
// Per tile:  D[k,b] = CB_tile(16k x 8d) * X^T(8d x 16b)  via 2x V_WMMA_F32_16X16X4_F32
// C/D layout: lane l, vgpr r  ->  k_local = r + 8*(l>>4),  b = l & 15
// => per-lane running argmin over k, one shfl_xor(16) to merge lane halves.

typedef __attribute__((ext_vector_type(2))) float v2f;
typedef __attribute__((ext_vector_type(4))) float v4f;
typedef __attribute__((ext_vector_type(8))) float v8f;

#define WAVES_PER_BLOCK 8
#define B_TOT   4096
#define C_TOT   128
#define K_TOT   256
#define D_EMB   8

__global__ __launch_bounds__(256) void vqvae_wmma_kernel(
    const float* __restrict__ x,          // (4096, 1024)
    const float* __restrict__ cb,         // (128, 256, 8)
    float* __restrict__ out_embed,        // (4096, 1024)
    float* __restrict__ out_onehot)       // (4096, 128, 256)
{
    __shared__ float lds_c2[WAVES_PER_BLOCK][K_TOT];   // ||codebook[c,k,:]||^2
    __shared__ int   lds_best[WAVES_PER_BLOCK][16];    // argmin per batch row

    const int lane = threadIdx.x & 31;
    const int w    = threadIdx.x >> 5;
    const int waveGlobal = blockIdx.x * WAVES_PER_BLOCK + w;   // 0..32767
    const int c    = waveGlobal & (C_TOT - 1);                 // code group
    const int b0   = (waveGlobal >> 7) << 4;                   // batch tile origin

    const int half = lane >> 4;     // 0: lanes 0-15, 1: lanes 16-31
    const int l16  = lane & 15;

    const float* __restrict__ cbc = cb + (size_t)c * K_TOT * D_EMB;

    // ---- Phase 1: codebook row squared norms -> LDS (8 rows per lane) ----
    #pragma unroll
    for (int j = 0; j < 8; ++j) {
        const int row = lane * 8 + j;                 // 32 lanes x 8 = 256 rows
        const float* r = cbc + row * D_EMB;
        v4f p0 = *(const v4f*)(r);
        v4f p1 = *(const v4f*)(r + 4);
        lds_c2[w][row] = p0.x*p0.x + p0.y*p0.y + p0.z*p0.z + p0.w*p0.w
                       + p1.x*p1.x + p1.y*p1.y + p1.z*p1.z + p1.w*p1.w;
    }
    // Wave-internal cross-lane RAW through LDS: drain DS counter.
    asm volatile("s_wait_dscnt 0x0" ::: "memory");

    // ---- Phase 2: B fragments (x^T tile, loop-invariant across k-tiles) ----
    // B is 4x16 (dxb). ISA B layout: row striped across lanes within a VGPR;
    // lanes 0-15 carry K=0,1 pairs, lanes 16-31 carry K=2,3 pairs -> float2 at d=2*half.
    const float* __restrict__ xrow = x + (size_t)(b0 + l16) * (C_TOT * D_EMB) + c * D_EMB;
    const v2f bf0 = *(const v2f*)(xrow + 2 * half);       // d = {0,1} or {2,3}
    const v2f bf1 = *(const v2f*)(xrow + 4 + 2 * half);   // d = {4,5} or {6,7}

    float bestVal = 3.4e38f;
    int   bestIdx = 0;

    // ---- Phase 3: 16 k-tiles of 16 codes; 2 WMMAs each (D=8 = 2 x K4) ----
    for (int t = 0; t < 16; ++t) {
        // A is 16x4 (kxd): lane m = codebook row, VGPR pair = K pairs (high lanes +2).
        const float* arow = cbc + (t * 16 + l16) * D_EMB;
        v2f af0 = *(const v2f*)(arow + 2 * half);
        v2f af1 = *(const v2f*)(arow + 4 + 2 * half);

        v8f acc = {};
        acc = __builtin_amdgcn_wmma_f32_16x16x4_f32(false, af0, false, bf0,
                                                    (short)0, acc, false, false);
        acc = __builtin_amdgcn_wmma_f32_16x16x4_f32(false, af1, false, bf1,
                                                    (short)0, acc, false, false);

        // score(k) = c2[k] - 2*dot  (x2 term is constant over k -> dropped)
        const int kbase = t * 16 + half * 8;
        v4f c2a = *(const v4f*)&lds_c2[w][kbase];
        v4f c2b = *(const v4f*)&lds_c2[w][kbase + 4];
        float c2v[8] = {c2a.x, c2a.y, c2a.z, c2a.w, c2b.x, c2b.y, c2b.z, c2b.w};
        #pragma unroll
        for (int r = 0; r < 8; ++r) {
            float s = c2v[r] - 2.0f * acc[r];
            int   k = kbase + r;
            if (s < bestVal) { bestVal = s; bestIdx = k; }   // ascending k => first-min
        }
    }

    // ---- Phase 4: merge the two lane halves (k%16<8 vs >=8) ----
    {
        float oVal = __shfl_xor(bestVal, 16, 32);
        int   oIdx = __shfl_xor(bestIdx, 16, 32);
        if (oVal < bestVal || (oVal == bestVal && oIdx < bestIdx)) {
            bestVal = oVal; bestIdx = oIdx;
        }
    }
    if (lane < 16) lds_best[w][lane] = bestIdx;
    asm volatile("s_wait_dscnt 0x0" ::: "memory");

    // ---- Phase 5: gather embeddings (16 rows, 32B each) ----
    if (lane < 16) {
        const float* src = cbc + bestIdx * D_EMB;
        v4f e0 = *(const v4f*)(src);
        v4f e1 = *(const v4f*)(src + 4);
        float* dst = out_embed + (size_t)(b0 + lane) * (C_TOT * D_EMB) + c * D_EMB;
        __builtin_nontemporal_store(e0, (v4f*)dst);
        __builtin_nontemporal_store(e1, (v4f*)(dst + 4));
    }

    // ---- Phase 6: one-hot tile, 16 rows x 256 k = 16KB, streamed NT ----
    // Composed branchlessly so a single store pass suffices (no ordering hazard).
    #pragma unroll 4
    for (int i = lane; i < 16 * (K_TOT / 4); i += 32) {
        const int row = i >> 6;            // 64 float4 per row
        const int q   = i & 63;
        const int kb  = q << 2;
        const int best = lds_best[w][row];
        v4f v;
        v.x = (best == kb    ) ? 1.0f : 0.0f;
        v.y = (best == kb + 1) ? 1.0f : 0.0f;
        v.z = (best == kb + 2) ? 1.0f : 0.0f;
        v.w = (best == kb + 3) ? 1.0f : 0.0f;
        float* dst = out_onehot + ((size_t)(b0 + row) * C_TOT + c) * K_TOT + kb;
        __builtin_nontemporal_store(v, (v4f*)dst);
    }
}

extern "C" void kernel_launch(void* const* d_in, const int* in_sizes, int n_in,
                              void* d_out, int out_size, void* d_ws, size_t ws_size,
                              hipStream_t stream) {
    (void)in_sizes; (void)n_in; (void)out_size; (void)d_ws; (void)ws_size;
    const float* x  = (const float*)d_in[0];
    const float* cb = (const float*)d_in[1];
    float* embed  = (float*)d_out;                          // 4096*1024
    float* onehot = (float*)d_out + (size_t)B_TOT * C_TOT * D_EMB;  // 4096*128*256

    // 32768 independent wave-tiles = 4096 blocks x 8 waves (256 threads, 8 wave32s)
    dim3 grid(4096), block(256);
    hipLaunchKernelGGL(vqvae_wmma_kernel, grid, block, 0, stream, x, cb, embed, onehot);
}